// SeparateTrajAndTrainableWeightedEmbeddingWithPos_74414603370570
// MI455X (gfx1250) — compile-verified
//
#include <hip/hip_runtime.h>
#include <math.h>

// Shapes from the reference
#define S_LEN 2048
#define BATCH 128
#define NKEY  64
#define DMODEL 256
#define TFEAT 6
#define OUTW  (TFEAT + DMODEL)   // 262
#define ROWS_PER_WG 64
#define SPAN 66                  // padded K-stride in LDS (conflict-free, 8B aligned)

#define LOG_2PI 1.8378770664093453f

typedef __attribute__((ext_vector_type(2))) float v2f;
typedef __attribute__((ext_vector_type(4))) float v4f;
typedef __attribute__((ext_vector_type(8))) float v8f;

__global__ __launch_bounds__(256) void fused_gauss_emb_kernel(
    const float* __restrict__ traj,    // (S,B,T)
    const float* __restrict__ xy,      // (S,B,2)
    const float* __restrict__ means,   // (K,2)
    const float* __restrict__ Amat,    // (K,2,2)
    const float* __restrict__ W,       // (D,K) row-major
    const float* __restrict__ bvec,    // (D,)
    const float* __restrict__ pe,      // (MAX_LEN,1,D)
    float* __restrict__ out)           // (S,B,T+D)
{
    __shared__ float s_keys[NKEY * 4];          // i00, i01, i11, logdet+2log2pi
    __shared__ float s_mx[NKEY];
    __shared__ float s_my[NKEY];
    __shared__ float s_lw[ROWS_PER_WG * SPAN];  // log-weight A tile, padded

    const int tid  = threadIdx.x;
    const int row0 = blockIdx.x * ROWS_PER_WG;  // global (s*B+b) row base
    const int sIdx = row0 >> 7;                 // same s for all 64 rows (64 | 128)

    // ---- per-key 2x2 Gaussian params -----------------------------------
    if (tid < NKEY) {
        float a00 = Amat[tid * 4 + 0], a01 = Amat[tid * 4 + 1];
        float a10 = Amat[tid * 4 + 2], a11 = Amat[tid * 4 + 3];
        float c00 = a00 * a00 + a01 * a01;
        float c01 = a00 * a10 + a01 * a11;
        float c11 = a10 * a10 + a11 * a11;
        float det = c00 * c11 - c01 * c01;
        float rdet = 1.0f / det;
        s_keys[tid * 4 + 0] = c11 * rdet;
        s_keys[tid * 4 + 1] = -c01 * rdet;
        s_keys[tid * 4 + 2] = c00 * rdet;
        s_keys[tid * 4 + 3] = __logf(det) + 2.0f * LOG_2PI;
        s_mx[tid] = means[tid * 2 + 0];
        s_my[tid] = means[tid * 2 + 1];
    }
    __syncthreads();

    // ---- log-weights A-tile: 64 rows x 64 keys into LDS ----------------
    {
        int m  = tid >> 2;              // 0..63 row within wg
        int kb = (tid & 3) << 4;        // 16 keys per thread
        int r  = row0 + m;
        float x = xy[r * 2 + 0];
        float y = xy[r * 2 + 1];
        #pragma unroll
        for (int j = 0; j < 16; ++j) {
            int k = kb + j;
            float dx = x - s_mx[k];
            float dy = y - s_my[k];
            const float* kp = &s_keys[k * 4];
            float quad = kp[0] * dx * dx + 2.0f * kp[1] * dx * dy + kp[2] * dy * dy;
            s_lw[m * SPAN + k] = -0.5f * (quad + kp[3]);
        }
    }
    __syncthreads();

    // ---- WMMA GEMM: (64 x 64) @ (64 x 256) via f32 16x16x4 chains ------
    const int wave  = tid >> 5;             // 8 waves
    const int lane  = tid & 31;
    const int mt    = (wave & 3) << 4;      // M-tile row offset: 0/16/32/48
    const int nh    = (wave >> 2) << 7;     // N half: 0 or 128 (8 tiles of 16)
    const int lrow  = lane & 15;
    const int khalf = (lane >> 4) << 1;     // lanes 16-31 hold K+2,K+3

    v8f acc[8] = {};                        // 8 accumulator tiles (64 VGPRs)
    const float* aptr = &s_lw[(mt + lrow) * SPAN + khalf];

    for (int kb = 0; kb < NKEY; kb += 4) {
        v2f a = *(const v2f*)(aptr + kb);   // A-frag: ds_load_b64
        #pragma unroll
        for (int t = 0; t < 8; ++t) {
            int ncol = nh + t * 16 + lrow;
            // B-frag (4x16, K x N): element (k, n) = W[n*K + k]; W is L2/WGP$-resident
            v2f bfrag = *(const v2f*)(W + ncol * NKEY + kb + khalf);
            acc[t] = __builtin_amdgcn_wmma_f32_16x16x4_f32(
                false, a, false, bfrag, (short)0, acc[t], false, false);
        }
    }

    // ---- epilogue: + b + pe, scatter per C/D VGPR layout ---------------
    const float* pes = pe + (size_t)sIdx * DMODEL;
    const int rbase = row0 + mt + ((lane >> 4) << 3);   // +8 for lanes 16-31
    #pragma unroll
    for (int t = 0; t < 8; ++t) {
        int d = nh + t * 16 + lrow;
        float add = bvec[d] + pes[d];
        #pragma unroll
        for (int v = 0; v < 8; ++v) {
            size_t r = (size_t)(rbase + v);
            out[r * OUTW + TFEAT + d] = acc[t][v] + add;
        }
    }

    // ---- copy traj_feats columns ---------------------------------------
    for (int i = tid; i < ROWS_PER_WG * TFEAT; i += 256) {
        int m = i / TFEAT;
        int c = i - m * TFEAT;
        size_t r = (size_t)(row0 + m);
        out[r * OUTW + c] = traj[r * TFEAT + c];
    }
}

extern "C" void kernel_launch(void* const* d_in, const int* in_sizes, int n_in,
                              void* d_out, int out_size, void* d_ws, size_t ws_size,
                              hipStream_t stream) {
    const float* traj  = (const float*)d_in[0];   // (S,B,T)
    const float* xy    = (const float*)d_in[1];   // (S,B,2)
    const float* means = (const float*)d_in[2];   // (K,2)
    const float* Amat  = (const float*)d_in[3];   // (K,2,2)
    const float* W     = (const float*)d_in[4];   // (D,K)
    const float* bvec  = (const float*)d_in[5];   // (D,)
    const float* pe    = (const float*)d_in[6];   // (MAX_LEN,1,D)
    float* out = (float*)d_out;

    const int total_rows = S_LEN * BATCH;                 // 262144
    dim3 grid(total_rows / ROWS_PER_WG);                  // 4096
    dim3 block(256);
    hipLaunchKernelGGL(fused_gauss_emb_kernel, grid, block, 0, stream,
                       traj, xy, means, Amat, W, bvec, pe, out);
}